// Lya_Spec_with_Jaco_74174085202263
// MI455X (gfx1250) — compile-verified
//
#include <hip/hip_runtime.h>
#include <stdint.h>

typedef unsigned int u32;
typedef uint64_t     u64;
typedef u32 v4u __attribute__((ext_vector_type(4)));
typedef int v4i __attribute__((ext_vector_type(4)));
typedef int v8i __attribute__((ext_vector_type(8)));

#define TSTEPS 2048
#define BATCH  8192
#define CH     8                 // timesteps per TDM chunk
#define ROWS   (CH * 3)          // 24 rows of 32 floats per chunk
#define NCHUNK (TSTEPS / CH)     // 256
#define NBUF   4                 // TDM pipeline depth (4 x 3KB LDS buffers)
#define BLOCK  32                // single-wave workgroups -> spread over WGPs

#define DT_C    0.01f
#define SIGMA_C 10.0f
#define RHO_C   28.0f
#define BETA_C  (8.0f / 3.0f)

// Issue one TDM 2-D tile load: ROWS rows (stride BATCH elems) x 32 floats -> LDS.
__device__ __forceinline__ void tdm_issue(u64 gaddr, u32 lds_byte, v8i g1) {
    v4u g0;
    g0.x = 1u;                                        // count=1, user descriptor
    g0.y = lds_byte;                                  // lds_addr (bytes)
    g0.z = (u32)gaddr;                                // global_addr[31:0]
    g0.w = ((u32)(gaddr >> 32) & 0x01FFFFFFu)         // global_addr[56:32]
           | (2u << 30);                              // type = 2 ("image")
    v4i gz = {0, 0, 0, 0};                            // groups 2/3 unused (2-D tile)
#if defined(__clang_major__) && (__clang_major__ >= 23)
    v8i gz8 = {0, 0, 0, 0, 0, 0, 0, 0};
    __builtin_amdgcn_tensor_load_to_lds(g0, g1, gz, gz, gz8, 0);
#else
    __builtin_amdgcn_tensor_load_to_lds(g0, g1, gz, gz, 0);
#endif
}

__global__ __launch_bounds__(BLOCK)
void lyap_tdm_kernel(const float* __restrict__ traj, float* __restrict__ out) {
    __shared__ float sh[NBUF * ROWS * 32];            // 12 KB: 4-deep per-wave pipeline

    const u32 lane = threadIdx.x & 31u;
    const u32 b0 = blockIdx.x * 32u;                  // this wave's first batch column

    u32 lds0 = __builtin_amdgcn_readfirstlane((u32)(uintptr_t)(void*)&sh[0]);
    const u64 gbase = (u64)(uintptr_t)traj + (u64)b0 * 4u;

    // D# group1 (invariant): data_size=4B; tensor 32 x ROWS; tile 32 x ROWS;
    // dim0 stride = BATCH elements (row index r = t*3 + var is contiguous).
    v8i g1;
    g1[0] = (2 << 16);           // data_size code 2 (4 bytes); no flags/pad
    g1[1] = (32 << 16);          // tensor_dim0[15:0] in bits[31:16]
    g1[2] = (ROWS << 16);        // tensor_dim0 hi=0 ; tensor_dim1[15:0]
    g1[3] = (32 << 16);          // tensor_dim1 hi=0 ; tile_dim0 = 32
    g1[4] = ROWS;                // tile_dim1 = ROWS ; tile_dim2 = 0
    g1[5] = BATCH;               // tensor_dim0_stride[31:0] = 8192 elements
    g1[6] = 0;
    g1[7] = 0;

    // Per-lane state: Q rows + log2-norm accumulators.
    float q00 = 1.f, q01 = 0.f, q02 = 0.f;
    float q10 = 0.f, q11 = 1.f, q12 = 0.f;
    float q20 = 0.f, q21 = 0.f, q22 = 1.f;
    float l0 = 0.f, l1 = 0.f, l2 = 0.f;

    // Prime the 4-deep TDM pipeline.
#pragma unroll
    for (int p = 0; p < NBUF; ++p)
        tdm_issue(gbase + (u64)(p * ROWS) * (u64)(BATCH * 4),
                  lds0 + (u32)(p * ROWS * 32 * 4), g1);

    for (int c = 0; c < NCHUNK; ++c) {
        // Ensure chunk c (oldest) is resident; keep up to 3 newer DMAs in flight.
        if (c + (NBUF - 1) < NCHUNK) __builtin_amdgcn_s_wait_tensorcnt(NBUF - 1);
        else                         __builtin_amdgcn_s_wait_tensorcnt(0);
        asm volatile("" ::: "memory");                // don't hoist LDS reads above wait

        const float* cb = &sh[(c & (NBUF - 1)) * (ROWS * 32)];
#pragma unroll
        for (int s = 0; s < CH; ++s) {
            const float x = cb[(s * 3 + 0) * 32 + lane];
            const float y = cb[(s * 3 + 1) * 32 + lane];
            const float z = cb[(s * 3 + 2) * 32 + lane];

            // J = I + DT * Jacobian(x,y,z); row 0 has a structural zero.
            const float a00 = 1.f - DT_C * SIGMA_C, a01 = DT_C * SIGMA_C;
            const float a10 = DT_C * (RHO_C - z),   a11 = 1.f - DT_C, a12 = -DT_C * x;
            const float a20 = DT_C * y,             a21 = DT_C * x,   a22 = 1.f - DT_C * BETA_C;

            // M = J * Q
            float m00 = a00 * q00 + a01 * q10;
            float m01 = a00 * q01 + a01 * q11;
            float m02 = a00 * q02 + a01 * q12;
            float m10 = a10 * q00 + a11 * q10 + a12 * q20;
            float m11 = a10 * q01 + a11 * q11 + a12 * q21;
            float m12 = a10 * q02 + a11 * q12 + a12 * q22;
            float m20 = a20 * q00 + a21 * q10 + a22 * q20;
            float m21 = a20 * q01 + a21 * q11 + a22 * q21;
            float m22 = a20 * q02 + a21 * q12 + a22 * q22;

            // Gram-Schmidt row 0:  log|r| = 0.5*ln2*log2(r.r); 1/|r| = rsq(r.r)
            float nn0  = m00 * m00 + m01 * m01 + m02 * m02;
            float inv0 = __builtin_amdgcn_rsqf(nn0);
            l0 += __builtin_amdgcn_logf(nn0);
            q00 = m00 * inv0; q01 = m01 * inv0; q02 = m02 * inv0;

            // row 1
            float d01 = q00 * m10 + q01 * m11 + q02 * m12;
            float r10 = m10 - d01 * q00;
            float r11 = m11 - d01 * q01;
            float r12 = m12 - d01 * q02;
            float nn1  = r10 * r10 + r11 * r11 + r12 * r12;
            float inv1 = __builtin_amdgcn_rsqf(nn1);
            l1 += __builtin_amdgcn_logf(nn1);
            q10 = r10 * inv1; q11 = r11 * inv1; q12 = r12 * inv1;

            // row 2
            float d02 = q00 * m20 + q01 * m21 + q02 * m22;
            float d12 = q10 * m20 + q11 * m21 + q12 * m22;
            float r20 = m20 - d02 * q00 - d12 * q10;
            float r21 = m21 - d02 * q01 - d12 * q11;
            float r22 = m22 - d02 * q02 - d12 * q12;
            float nn2  = r20 * r20 + r21 * r21 + r22 * r22;
            float inv2 = __builtin_amdgcn_rsqf(nn2);
            l2 += __builtin_amdgcn_logf(nn2);
            q20 = r20 * inv2; q21 = r21 * inv2; q22 = r22 * inv2;
        }
        asm volatile("" ::: "memory");                // LDS reads complete before re-issue

        // Refill the buffer we just consumed with chunk c+NBUF.
        if (c + NBUF < NCHUNK)
            tdm_issue(gbase + (u64)((c + NBUF) * ROWS) * (u64)(BATCH * 4),
                      lds0 + (u32)((c & (NBUF - 1)) * ROWS * 32 * 4), g1);
    }

    // sum log|r| / (T*DT); accumulators hold log2(nn) = 2*log2|r|.
    const float scale = 0.5f * 0.69314718055994530942f / (TSTEPS * DT_C);
    const u32 b = b0 + lane;
    out[0 * BATCH + b] = l0 * scale;
    out[1 * BATCH + b] = l1 * scale;
    out[2 * BATCH + b] = l2 * scale;
}

extern "C" void kernel_launch(void* const* d_in, const int* in_sizes, int n_in,
                              void* d_out, int out_size, void* d_ws, size_t ws_size,
                              hipStream_t stream) {
    (void)in_sizes; (void)n_in; (void)d_ws; (void)ws_size; (void)out_size;
    const float* traj = (const float*)d_in[0];
    float* out = (float*)d_out;
    dim3 grid(BATCH / 32);
    dim3 block(BLOCK);
    hipLaunchKernelGGL(lyap_tdm_kernel, grid, block, 0, stream, traj, out);
}